// CustomFullyConnectedLayerGoogleTopK2_65618510348679
// MI455X (gfx1250) — compile-verified
//
#include <hip/hip_runtime.h>
#include <stdint.h>

// ---------------------------------------------------------------------------
// Problem constants (from reference): n = 4096, K = ceil(0.01*4096) = 41,
// l = 0.01, 50 Dykstra iterations, x is (1024, 4096) fp32, V is (4096,4096).
// Workspace layout: [0, 16KB) -> mask m (4096 f32), [16KB, 16KB+64MB) -> WT.
// ---------------------------------------------------------------------------

typedef __attribute__((ext_vector_type(2))) float v2f;
typedef __attribute__((ext_vector_type(8))) float v8f;

#define N_FEAT   4096
#define K_SUM    41.0f
#define BATCH    1024

// ----------------------------- async helpers -------------------------------
__device__ __forceinline__ void async_ld_b128(unsigned lds_byte, const void* g) {
  asm volatile("global_load_async_to_lds_b128 %0, %1, off"
               :: "v"(lds_byte), "v"(g) : "memory");
}
__device__ __forceinline__ void async_ld_b32(unsigned lds_byte, const void* g) {
  asm volatile("global_load_async_to_lds_b32 %0, %1, off"
               :: "v"(lds_byte), "v"(g) : "memory");
}
__device__ __forceinline__ void wait_async0() {
  asm volatile("s_wait_asynccnt 0" ::: "memory");
}
__device__ __forceinline__ unsigned lds_addr_of(const void* p) {
  // LDS aperture: addr[31:0] is the LDS byte offset (ISA 10.2 aperture calc).
  return (unsigned)(uintptr_t)p;
}

// ---------------------------------------------------------------------------
// Kernel 1: Dykstra soft-top-k mask.  The hyperplane residual p is uniform
// (p_new = (sum(yp)-k)/n for every element), so p is a scalar; only y and q
// are per-element.  One block, 256 threads x 16 elements.
// ---------------------------------------------------------------------------
__global__ __launch_bounds__(256) void dykstra_mask_kernel(
    const float* __restrict__ alpha, float* __restrict__ m_out) {
  __shared__ float red[256];
  const int tid = threadIdx.x;
  float y[16], q[16];
  float p = 0.0f;
#pragma unroll
  for (int e = 0; e < 16; ++e) {
    y[e] = alpha[tid + 256 * e] * 100.0f;   // x / l,  l = 0.01
    q[e] = 0.0f;
  }
  for (int it = 0; it < 50; ++it) {
    float s = 0.0f;
#pragma unroll
    for (int e = 0; e < 16; ++e) { y[e] += p; s += y[e]; }   // yp
    red[tid] = s;
    __syncthreads();
#pragma unroll
    for (int off = 128; off >= 1; off >>= 1) {
      if (tid < off) red[tid] += red[tid + off];
      __syncthreads();
    }
    const float tot = red[0];
    __syncthreads();                         // protect red before next write
    const float delta = (tot - K_SUM) * (1.0f / (float)N_FEAT);
    p = delta;                               // p_new = yp - y_hp = delta
#pragma unroll
    for (int e = 0; e < 16; ++e) {
      const float yq = (y[e] - delta) + q[e];
      const float yb = fminf(fmaxf(yq, 0.0f), 1.0f);
      q[e] = yq - yb;
      y[e] = yb;
    }
  }
#pragma unroll
  for (int e = 0; e < 16; ++e) m_out[tid + 256 * e] = y[e];
}

// ---------------------------------------------------------------------------
// Kernel 2: WT[c][r] = m[i] * V[i][c] with i = (r - c) mod n  (so B[k=c][n=r]
// for the GEMM).  64x64 tiles through LDS so that V reads AND WT writes are
// both coalesced (WT is written along 64-long contiguous r-runs).
// ---------------------------------------------------------------------------
__global__ __launch_bounds__(256) void build_wt_kernel(
    const float* __restrict__ V, const float* __restrict__ m,
    float* __restrict__ WT) {
  __shared__ float tile[64 * 65];   // stride 65 -> conflict-free column reads
  __shared__ float msh[64];
  const int i0 = blockIdx.x * 64;   // diagonal-id tile
  const int c0 = blockIdx.y * 64;   // column tile
  const int t  = threadIdx.x;
  if (t < 64) msh[t] = m[i0 + t];
#pragma unroll
  for (int it = 0; it < 16; ++it) {
    const int row = (t >> 6) + it * 4;     // 0..63  (diagonal id - i0)
    const int cl  = t & 63;                // 0..63  (column - c0)
    tile[row * 65 + cl] = V[(size_t)(i0 + row) * N_FEAT + c0 + cl];
  }
  __syncthreads();
#pragma unroll
  for (int it = 0; it < 16; ++it) {
    const int cl = (t >> 6) + it * 4;
    const int j  = t & 63;                 // lanes sweep j -> contiguous r
    const int c  = c0 + cl;
    const int r  = (c + i0 + j) & (N_FEAT - 1);
    WT[(size_t)c * N_FEAT + r] = tile[j * 65 + cl] * msh[j];
  }
}

// ---------------------------------------------------------------------------
// Kernel 3: y(1024x4096) = x(1024x4096) @ WT(4096x4096), fp32 WMMA.
// Block tile 128x128, BK=32, 8 waves; wave tile 32(M) x 64(N) = 2x4 C tiles.
// A in LDS row-major [128][36], B in LDS n-major [128][36] (stride 36 floats:
// 144B rows -> 16B aligned for b128 async writes; 36 = 4*9, 9 coprime 16 ->
// conflict-free b64 fragment reads).  Double-buffered async-to-LDS.
// ---------------------------------------------------------------------------
#define BM   128
#define BN   128
#define BK   32
#define LDA  36
#define ABUF (128 * 36)     // floats per buffer
#define NCHUNK (N_FEAT / BK)

__global__ __launch_bounds__(256) void wmma_gemm_kernel(
    const float* __restrict__ X, const float* __restrict__ WT,
    float* __restrict__ Y) {
  __shared__ float lds[4 * ABUF];                 // 72 KB
  const int t     = threadIdx.x;
  const int n0    = blockIdx.x * BN;
  const int m0    = blockIdx.y * BM;
  const int lane  = t & 31;
  const int wid   = t >> 5;                       // 0..7
  const int wm    = (wid >> 1) * 32;              // wave M offset in block
  const int wn    = (wid & 1) * 64;               // wave N offset in block
  const int mloc  = lane & 15;                    // M (or N) within 16-tile
  const int khalf = lane >> 4;                    // K-half select (ISA layout)

  float* const Ab0 = lds;
  float* const Bb0 = lds + 2 * ABUF;
  const unsigned aL0 = lds_addr_of(Ab0);
  const unsigned bL0 = lds_addr_of(Bb0);

  // --- per-thread load cursors (advance by BK in K each chunk) -------------
  const float* ga[4];
  unsigned     la[4];
#pragma unroll
  for (int i = 0; i < 4; ++i) {
    const int s   = t + 256 * i;        // 1024 b128 slots
    const int row = s >> 3;             // 0..127
    const int c4  = (s & 7) * 4;        // 0,4,...,28
    ga[i] = X + (size_t)(m0 + row) * N_FEAT + c4;
    la[i] = (unsigned)(row * LDA + c4) * 4u;
  }
  const int kk0 = t >> 7;               // 0..1
  const int nn  = t & 127;              // 0..127
  const float* gb = WT + (size_t)kk0 * N_FEAT + n0 + nn;
  const unsigned lb = (unsigned)(nn * LDA + kk0) * 4u;

  auto issue = [&](int buf) {
    const unsigned aL = aL0 + (unsigned)buf * (ABUF * 4u);
    const unsigned bL = bL0 + (unsigned)buf * (ABUF * 4u);
#pragma unroll
    for (int i = 0; i < 4; ++i) {                 // A: 4 x b128 per thread
      async_ld_b128(aL + la[i], ga[i]);
      ga[i] += BK;
    }
#pragma unroll
    for (int i = 0; i < 16; ++i) {                // B: 16 x b32 per thread
      async_ld_b32(bL + lb + (unsigned)i * 8u, gb + (size_t)i * 2 * N_FEAT);
    }
    gb += (size_t)BK * N_FEAT;
  };

  v8f acc[2][4];
#pragma unroll
  for (int i = 0; i < 2; ++i)
#pragma unroll
    for (int j = 0; j < 4; ++j)
      acc[i][j] = v8f{0.f, 0.f, 0.f, 0.f, 0.f, 0.f, 0.f, 0.f};

  issue(0);
  wait_async0();
  __syncthreads();

  for (int ch = 0; ch < NCHUNK; ++ch) {
    const int buf = ch & 1;
    if (ch + 1 < NCHUNK) issue(buf ^ 1);          // prefetch next chunk
    const float* As = Ab0 + buf * ABUF;
    const float* Bs = Bb0 + buf * ABUF;
#pragma unroll
    for (int kk = 0; kk < BK; kk += 4) {
      const int ko = kk + 2 * khalf;
      v2f a0 = *(const v2f*)&As[(wm +      mloc) * LDA + ko];
      v2f a1 = *(const v2f*)&As[(wm + 16 + mloc) * LDA + ko];
      v2f b0 = *(const v2f*)&Bs[(wn +      mloc) * LDA + ko];
      v2f b1 = *(const v2f*)&Bs[(wn + 16 + mloc) * LDA + ko];
      v2f b2 = *(const v2f*)&Bs[(wn + 32 + mloc) * LDA + ko];
      v2f b3 = *(const v2f*)&Bs[(wn + 48 + mloc) * LDA + ko];
      acc[0][0] = __builtin_amdgcn_wmma_f32_16x16x4_f32(false, a0, false, b0, (short)0, acc[0][0], false, false);
      acc[0][1] = __builtin_amdgcn_wmma_f32_16x16x4_f32(false, a0, false, b1, (short)0, acc[0][1], false, false);
      acc[0][2] = __builtin_amdgcn_wmma_f32_16x16x4_f32(false, a0, false, b2, (short)0, acc[0][2], false, false);
      acc[0][3] = __builtin_amdgcn_wmma_f32_16x16x4_f32(false, a0, false, b3, (short)0, acc[0][3], false, false);
      acc[1][0] = __builtin_amdgcn_wmma_f32_16x16x4_f32(false, a1, false, b0, (short)0, acc[1][0], false, false);
      acc[1][1] = __builtin_amdgcn_wmma_f32_16x16x4_f32(false, a1, false, b1, (short)0, acc[1][1], false, false);
      acc[1][2] = __builtin_amdgcn_wmma_f32_16x16x4_f32(false, a1, false, b2, (short)0, acc[1][2], false, false);
      acc[1][3] = __builtin_amdgcn_wmma_f32_16x16x4_f32(false, a1, false, b3, (short)0, acc[1][3], false, false);
    }
    wait_async0();
    __syncthreads();
  }

  // C/D layout (ISA 7.12.2): vgpr g holds M = g + 8*khalf, N = mloc.
#pragma unroll
  for (int i = 0; i < 2; ++i)
#pragma unroll
    for (int j = 0; j < 4; ++j)
#pragma unroll
      for (int g = 0; g < 8; ++g) {
        const int row = m0 + wm + i * 16 + g + 8 * khalf;
        const int col = n0 + wn + j * 16 + mloc;
        Y[(size_t)row * N_FEAT + col] = acc[i][j][g];
      }
}

// ---------------------------------------------------------------------------
extern "C" void kernel_launch(void* const* d_in, const int* in_sizes, int n_in,
                              void* d_out, int out_size, void* d_ws, size_t ws_size,
                              hipStream_t stream) {
  (void)in_sizes; (void)n_in; (void)out_size; (void)ws_size;
  const float* x     = (const float*)d_in[0];   // (1024, 4096) f32
  const float* V     = (const float*)d_in[1];   // (4096, 4096) f32
  const float* alpha = (const float*)d_in[2];   // (4096,)      f32
  float* y  = (float*)d_out;                    // (1024, 4096) f32
  float* m  = (float*)d_ws;                                    // 16 KB
  float* WT = (float*)((char*)d_ws + 16384);                   // 64 MB

  dykstra_mask_kernel<<<1, 256, 0, stream>>>(alpha, m);
  build_wt_kernel<<<dim3(N_FEAT / 64, N_FEAT / 64), 256, 0, stream>>>(V, m, WT);
  wmma_gemm_kernel<<<dim3(N_FEAT / BN, BATCH / BM), 256, 0, stream>>>(x, WT, y);
}